// LSTM_57604101374649
// MI455X (gfx1250) — compile-verified
//
#include <hip/hip_runtime.h>
#include <hip/hip_bf16.h>

// ---------------------------------------------------------------------------
// Grid-cell LSTM forward, fp32, CDNA5 (gfx1250) using V_WMMA_F32_16X16X4_F32.
// B=256, T=100, Ng=2048, Np=512.  All matrix math on the fp32 WMMA path.
// ---------------------------------------------------------------------------

#define NG   2048
#define NP   512
#define TT   100
#define BB   256
#define G4   (4 * NG)   // 8192

typedef __attribute__((ext_vector_type(2))) float v2f;
typedef __attribute__((ext_vector_type(8))) float v8f;

static __device__ __forceinline__ v8f wmma_f32_k4(v2f a, v2f b, v8f c) {
    // D = A(16x4) * B(4x16) + C(16x16), fp32 in/out.
    return __builtin_amdgcn_wmma_f32_16x16x4_f32(
        /*neg_a=*/false, a, /*neg_b=*/false, b,
        /*c_mod=*/(short)0, c, /*reuse_a=*/false, /*reuse_b=*/false);
}

static __device__ __forceinline__ float sigmoidf_fast(float x) {
    return 1.0f / (1.0f + __expf(-x));
}

// ---------------------------------------------------------------------------
// Kernel 1: fold the input projection.
//   Wc[0][n] = sum_k M_W[0][k]  * lstm_W[k][n]
//   Wc[1][n] = sum_k M_W[1][k]  * lstm_W[k][n]
//   bc[n]    = sum_k M_b[k]     * lstm_W[k][n] + lstm_b[n]
// lstm_W reads are fully coalesced (threads stride over n).
// ---------------------------------------------------------------------------
__global__ void fold_input_proj(const float* __restrict__ M_W,
                                const float* __restrict__ M_b,
                                const float* __restrict__ lstm_W,
                                const float* __restrict__ lstm_b,
                                float* __restrict__ Wc,
                                float* __restrict__ bc) {
    const int n = blockIdx.x * 256 + threadIdx.x;   // 0 .. 8191
    float s0 = 0.f, s1 = 0.f, sb = 0.f;
    for (int k = 0; k < NG; ++k) {
        const float w = lstm_W[(size_t)k * G4 + n];
        s0 = fmaf(M_W[k],      w, s0);   // uniform scalar operand -> SMEM path
        s1 = fmaf(M_W[NG + k], w, s1);
        sb = fmaf(M_b[k],      w, sb);
    }
    Wc[n]       = s0;
    Wc[G4 + n]  = s1;
    bc[n]       = sb + lstm_b[n];
}

// ---------------------------------------------------------------------------
// Kernel 2: generic fp32 WMMA GEMM with bias (+optional ReLU).
//   C[M,N] = act(A[M,K] @ B[K,N] + bias[N])
// Block = 256 threads = 8 waves; each wave owns a 64x64 C tile
// (16 accumulators of v8f).  Block footprint: 128(M) x 256(N).
// Requires M%128==0, N%256==0, K%4==0 (true for all uses here).
// ---------------------------------------------------------------------------
template <bool RELU>
__global__ __launch_bounds__(256)
void gemm_bias_wmma(const float* __restrict__ A, int lda,
                    const float* __restrict__ B, int ldb,
                    const float* __restrict__ bias,
                    float* __restrict__ C, int ldc,
                    int K) {
    const int lane = threadIdx.x & 31;
    const int wave = threadIdx.x >> 5;
    const int lh   = lane >> 4;     // 0: lanes 0-15, 1: lanes 16-31
    const int ll   = lane & 15;

    const int mb = blockIdx.y * 128 + (wave >> 2) * 64;
    const int nb = blockIdx.x * 256 + (wave & 3) * 64;

    v8f acc[4][4];
#pragma unroll
    for (int mt = 0; mt < 4; ++mt)
#pragma unroll
        for (int nt = 0; nt < 4; ++nt) acc[mt][nt] = (v8f)0.f;

    for (int k = 0; k < K; k += 4) {
        v2f afrag[4];
#pragma unroll
        for (int mt = 0; mt < 4; ++mt) {
            // A 16x4 frag: lane ll holds row mb+mt*16+ll, cols {k+2*lh, k+2*lh+1}
            afrag[mt] = *(const v2f*)(A + (size_t)(mb + mt * 16 + ll) * lda
                                        + k + 2 * lh);
        }
        v2f bfrag[4];
#pragma unroll
        for (int nt = 0; nt < 4; ++nt) {
            // B 4x16 frag: lane holds col nb+nt*16+ll, rows {k+2*lh, k+2*lh+1}
            const float* bp = B + (size_t)(k + 2 * lh) * ldb + nb + nt * 16 + ll;
            bfrag[nt] = (v2f){bp[0], bp[ldb]};
        }
#pragma unroll
        for (int mt = 0; mt < 4; ++mt)
#pragma unroll
            for (int nt = 0; nt < 4; ++nt)
                acc[mt][nt] = wmma_f32_k4(afrag[mt], bfrag[nt], acc[mt][nt]);
    }

#pragma unroll
    for (int mt = 0; mt < 4; ++mt)
#pragma unroll
        for (int nt = 0; nt < 4; ++nt) {
            const int n = nb + nt * 16 + ll;
            const float bn = bias[n];
#pragma unroll
            for (int r = 0; r < 8; ++r) {
                const int m = mb + mt * 16 + r + 8 * lh;
                float val = acc[mt][nt][r] + bn;
                if (RELU) val = fmaxf(val, 0.f);
                C[(size_t)m * ldc + n] = val;
            }
        }
}

// ---------------------------------------------------------------------------
// Kernel 3: one fused LSTM step.
//   z = xt + h_prev @ lstm_U        (xt folded into accumulator init)
//   i,f = sigmoid; g = relu; o = sigmoid   (gate order i,f,g,o)
//   c = f*c + i*g ;  h = o*relu(c)
// Each wave owns a 32(M) x 16(n') tile replicated across the 4 gate column
// bands of lstm_U, so the gate combine is purely in-register.
// Grid: x = 2048/(8 waves*16) = 16, y = 256/32 = 8  -> 1024 waves/step.
// lstm_U (64 MB) is L2-resident across the 100 sequential steps.
// ---------------------------------------------------------------------------
__global__ __launch_bounds__(256)
void lstm_step_wmma(const float* __restrict__ Aprev, int lda,   // h_{t-1}
                    const float* __restrict__ U,                // [2048, 8192]
                    const float* __restrict__ Wc,               // [2, 8192]
                    const float* __restrict__ bc,               // [8192]
                    const float* __restrict__ vin,              // [B, T, 2]
                    float* __restrict__ cbuf,                   // [B, 2048]
                    float* __restrict__ hs,                     // [B, T, 2048]
                    int t) {
    const int lane = threadIdx.x & 31;
    const int wave = threadIdx.x >> 5;
    const int lh   = lane >> 4;
    const int ll   = lane & 15;

    const int mb = blockIdx.y * 32;                       // batch-row base
    const int np = blockIdx.x * 128 + wave * 16 + ll;     // gate-relative col

    // Folded input projection coefficients for this lane's column, per gate.
    float w0[4], w1[4], bg[4];
#pragma unroll
    for (int g = 0; g < 4; ++g) {
        const int ng = np + g * NG;
        w0[g] = Wc[ng];
        w1[g] = Wc[G4 + ng];
        bg[g] = bc[ng];
    }

    // acc[gate][mt] initialized with xt = vx*w0 + vy*w1 + bc.
    v8f acc[4][2];
#pragma unroll
    for (int mt = 0; mt < 2; ++mt)
#pragma unroll
        for (int r = 0; r < 8; ++r) {
            const int m = mb + mt * 16 + r + 8 * lh;
            const v2f vv = *(const v2f*)(vin + ((size_t)m * TT + t) * 2);
#pragma unroll
            for (int g = 0; g < 4; ++g)
                acc[g][mt][r] = fmaf(vv.x, w0[g], fmaf(vv.y, w1[g], bg[g]));
        }

    // Recurrent GEMM: z += h_prev @ U over K = 2048.
    for (int k = 0; k < NG; k += 4) {
        v2f afrag[2];
#pragma unroll
        for (int mt = 0; mt < 2; ++mt)
            afrag[mt] = *(const v2f*)(Aprev + (size_t)(mb + mt * 16 + ll) * lda
                                            + k + 2 * lh);
        v2f bfrag[4];
#pragma unroll
        for (int g = 0; g < 4; ++g) {
            const float* bp = U + (size_t)(k + 2 * lh) * G4 + np + g * NG;
            bfrag[g] = (v2f){bp[0], bp[G4]};
        }
#pragma unroll
        for (int g = 0; g < 4; ++g)
#pragma unroll
            for (int mt = 0; mt < 2; ++mt)
                acc[g][mt] = wmma_f32_k4(afrag[mt], bfrag[g], acc[g][mt]);
    }

    // Gate math + state update (each (m,n') element owned by exactly one lane).
#pragma unroll
    for (int mt = 0; mt < 2; ++mt)
#pragma unroll
        for (int r = 0; r < 8; ++r) {
            const int m = mb + mt * 16 + r + 8 * lh;
            const size_t ci = (size_t)m * NG + np;
            const float ig = sigmoidf_fast(acc[0][mt][r]);
            const float fg = sigmoidf_fast(acc[1][mt][r]);
            const float gg = fmaxf(acc[2][mt][r], 0.f);
            const float og = sigmoidf_fast(acc[3][mt][r]);
            const float cn = fmaf(fg, cbuf[ci], ig * gg);
            cbuf[ci] = cn;
            hs[((size_t)m * TT + t) * NG + np] = og * fmaxf(cn, 0.f);
        }
}

// ---------------------------------------------------------------------------
// Host-side orchestration (graph-capture safe: only kernel launches).
// ---------------------------------------------------------------------------
extern "C" void kernel_launch(void* const* d_in, const int* in_sizes, int n_in,
                              void* d_out, int out_size, void* d_ws, size_t ws_size,
                              hipStream_t stream) {
    (void)in_sizes; (void)n_in; (void)out_size; (void)ws_size;

    const float* v       = (const float*)d_in[0];   // [B, T, 2]
    const float* p0      = (const float*)d_in[1];   // [B, Np]
    const float* enc1_W  = (const float*)d_in[2];   // [Np, Ng]
    const float* enc1_b  = (const float*)d_in[3];
    const float* enc2_W  = (const float*)d_in[4];
    const float* enc2_b  = (const float*)d_in[5];
    const float* M_W     = (const float*)d_in[6];   // [2, Ng]
    const float* M_b     = (const float*)d_in[7];   // [Ng]
    const float* lstm_W  = (const float*)d_in[8];   // [Ng, 4Ng]
    const float* lstm_U  = (const float*)d_in[9];   // [Ng, 4Ng]
    const float* lstm_b  = (const float*)d_in[10];  // [4Ng]
    const float* dense_W = (const float*)d_in[11];  // [Ng, Ng]
    const float* dense_b = (const float*)d_in[12];
    const float* dec_W   = (const float*)d_in[13];  // [Ng, Np]
    const float* dec_b   = (const float*)d_in[14];
    float* out = (float*)d_out;                     // [B, T, Np]

    // Workspace layout (floats).
    float* ws  = (float*)d_ws;
    float* Wc  = ws;                          // 2 * 8192
    float* bc  = Wc + 2 * G4;                 // 8192
    float* h0  = bc + G4;                     // 256 * 2048
    float* cbf = h0 + (size_t)BB * NG;        // 256 * 2048
    float* hs  = cbf + (size_t)BB * NG;       // 256 * 100 * 2048  (~210 MB)
    float* gcl = hs + (size_t)BB * TT * NG;   // 256 * 100 * 2048  (~210 MB)

    // 1) Fold input projection: Wc = M_W @ lstm_W, bc = M_b @ lstm_W + lstm_b.
    fold_input_proj<<<G4 / 256, 256, 0, stream>>>(M_W, M_b, lstm_W, lstm_b, Wc, bc);

    // 2) Initial state: h0 = p0 @ enc1_W + enc1_b, c0 = p0 @ enc2_W + enc2_b.
    gemm_bias_wmma<false><<<dim3(NG / 256, BB / 128), 256, 0, stream>>>(
        p0, NP, enc1_W, NG, enc1_b, h0, NG, NP);
    gemm_bias_wmma<false><<<dim3(NG / 256, BB / 128), 256, 0, stream>>>(
        p0, NP, enc2_W, NG, enc2_b, cbf, NG, NP);

    // 3) 100 sequential LSTM steps; h history lives in hs[b][t][n].
    for (int t = 0; t < TT; ++t) {
        const float* Ap = (t == 0) ? h0 : (hs + (size_t)(t - 1) * NG);
        const int lda   = (t == 0) ? NG : TT * NG;   // row stride of h_{t-1}
        lstm_step_wmma<<<dim3(NG / 128, BB / 32), 256, 0, stream>>>(
            Ap, lda, lstm_U, Wc, bc, v, cbf, hs, t);
    }

    // 4) g_cells = relu(hs @ dense_W + dense_b)   [25600 x 2048]
    gemm_bias_wmma<true><<<dim3(NG / 256, (BB * TT) / 128), 256, 0, stream>>>(
        hs, NG, dense_W, NG, dense_b, gcl, NG, NG);

    // 5) place_preds = g_cells @ dec_W + dec_b    [25600 x 512]
    gemm_bias_wmma<false><<<dim3(NP / 256, (BB * TT) / 128), 256, 0, stream>>>(
        gcl, NG, dec_W, NP, dec_b, out, NP, NG);
}